// Interpolate_47785806135531
// MI455X (gfx1250) — compile-verified
//
#include <hip/hip_runtime.h>

// Trilinear interpolation gather for MI455X (gfx1250).
//
// Shape facts: grid [16,128,128,128] f32 = 134 MB -> resident in the 192 MB L2.
// 1M random points * 16 ch * 8 corners. Gather-bound; WMMA does not apply
// (the contracted operand varies in both free indices -> no shared matrix).
// Strategy: point-per-thread wave32 gathers, pair-loads (b64) along the
// innermost dim, NT hints on streaming data so the grid stays L2-resident.

namespace {

constexpr int GD  = 128;                       // spatial extent per dim
constexpr int NCH = 16;                        // channels
constexpr long long CH_STRIDE = (long long)GD * GD * GD;  // elems per channel

__global__ __launch_bounds__(256) void trilerp16_kernel(
    const float* __restrict__ grid,
    const float* __restrict__ coords,
    float* __restrict__ out,
    int npts)
{
    int p = blockIdx.x * blockDim.x + threadIdx.x;
    if (p >= npts) return;

    // Prefetch the coord stream well ahead (gfx1250 global_prefetch_b8).
    // Speculative: out-of-range translations are silently dropped.
    __builtin_prefetch(coords + 3ll * (p + 16384), 0, 0);

    // Streaming (use-once) coord loads: non-temporal so the 12 MB coord
    // stream does not evict the L2-resident 134 MB grid.
    float cx = __builtin_nontemporal_load(coords + 3ll * p + 0);
    float cy = __builtin_nontemporal_load(coords + 3ll * p + 1);
    float cz = __builtin_nontemporal_load(coords + 3ll * p + 2);

    // Map [-1,1] -> [0, 127]; clamp for fp-rounding edge cases.
    const float scale = 0.5f * (float)(GD - 1);   // 63.5
    float gx = fminf(fmaxf((cx + 1.0f) * scale, 0.0f), (float)(GD - 1));
    float gy = fminf(fmaxf((cy + 1.0f) * scale, 0.0f), (float)(GD - 1));
    float gz = fminf(fmaxf((cz + 1.0f) * scale, 0.0f), (float)(GD - 1));

    // i in [0,126]; f in [0,1]. At g==127 this gives i=126,f=1 which exactly
    // reproduces the reference's index-clamp semantics (weight moves to the
    // hi corner) while keeping (k,k+1) contiguous -> always-legal b64 pairs.
    int   i  = min((int)gx, GD - 2);
    int   j  = min((int)gy, GD - 2);
    int   k  = min((int)gz, GD - 2);
    float fi = gx - (float)i;
    float fj = gy - (float)j;
    float fk = gz - (float)k;

    // Per-(i,j) corner blend weights; innermost dim handled by the lerp.
    float a00 = (1.0f - fi) * (1.0f - fj);
    float a01 = (1.0f - fi) * fj;
    float a10 = fi * (1.0f - fj);
    float a11 = fi * fj;
    float w0  = 1.0f - fk;
    float w1  = fk;

    // Base element offset inside one channel; the other three pair bases are
    // +GD, +GD*GD, +GD*GD+GD elements -> fold into 24-bit IOFFSET of the loads.
    int e = (i * GD + j) * GD + k;
    const float* gp = grid + e;

    long long pll = p;

    #pragma unroll
    for (int c = 0; c < NCH; ++c) {
        // Four b64 gathers = the 8 corners of this channel's cell.
        float2 d00 = *(const float2*)(gp);
        float2 d01 = *(const float2*)(gp + GD);
        float2 d10 = *(const float2*)(gp + GD * GD);
        float2 d11 = *(const float2*)(gp + GD * GD + GD);

        float r = a00 * (d00.x * w0 + d00.y * w1)
                + a01 * (d01.x * w0 + d01.y * w1)
                + a10 * (d10.x * w0 + d10.y * w1)
                + a11 * (d11.x * w0 + d11.y * w1);

        // Output is 64 MB write-once: non-temporal store keeps it from
        // evicting the grid out of L2. Coalesced per channel.
        __builtin_nontemporal_store(r, out + (long long)c * npts + pll);

        gp += CH_STRIDE;   // next channel, +8 MB
    }
}

} // namespace

extern "C" void kernel_launch(void* const* d_in, const int* in_sizes, int n_in,
                              void* d_out, int out_size, void* d_ws, size_t ws_size,
                              hipStream_t stream) {
    const float* grid   = (const float*)d_in[0];   // [16,128,128,128] f32
    const float* coords = (const float*)d_in[1];   // [N,3] f32 in [-1,1]
    float*       out    = (float*)d_out;           // [16,N] f32

    int npts = in_sizes[1] / 3;
    int threads = 256;                             // 8 wave32 per block
    int blocks  = (npts + threads - 1) / threads;

    trilerp16_kernel<<<blocks, threads, 0, stream>>>(grid, coords, out, npts);
}